// SimpleGCN_69784628626125
// MI455X (gfx1250) — compile-verified
//
#include <hip/hip_runtime.h>
#include <math.h>

#define D 128
#define BPITCH 144  // pairs per j-slice; 2*BPITCH % 64 == 32 -> wave halves use disjoint LDS banks
#define GEMM_LDS_BYTES (64 * BPITCH * 2 * (int)sizeof(float))  // 73,728 B

typedef float v2f __attribute__((ext_vector_type(2)));
typedef float v8f __attribute__((ext_vector_type(8)));

// ---------------- degree / norm ----------------
__global__ void k_deg_init(float* deg, int N) {
    int i = blockIdx.x * blockDim.x + threadIdx.x;
    if (i < N) deg[i] = 1.0f;  // self-loop
}

__global__ void k_deg_count(const int* __restrict__ dst, float* deg, int E) {
    int e = blockIdx.x * blockDim.x + threadIdx.x;
    if (e < E) atomicAdd(&deg[dst[e]], 1.0f);
}

__global__ void k_rsqrt(float* deg, int N) {
    int i = blockIdx.x * blockDim.x + threadIdx.x;
    if (i < N) deg[i] = rsqrtf(deg[i]);
}

// ---------------- dense GEMM: OUT[M x 128] = X[M x 128] @ W[128 x 128] ----------------
// W staged once per block into LDS in WMMA B-fragment order:
//   wl[(j*BPITCH + n)*2 .. +1] = { W[2j][n], W[2j+1][n] },  j = K-pair index 0..63
// Each wave computes a 16-row x 128-col strip (8 tiles), K stepped by 4 with
// v_wmma_f32_16x16x4_f32. Per K-step: 1 global b64 (A) + 4 ds_load_2addr_b64 into
// distinct regs (all 8 B fragments live), then 8 back-to-back WMMAs.
__global__ __launch_bounds__(256)
void k_gemm_wmma(const float* __restrict__ X, const float* __restrict__ W,
                 float* __restrict__ OUT, int M) {
    extern __shared__ float wl[];

    // cooperative stage: coalesced global reads, conflict-free LDS b64 writes
    for (int idx = threadIdx.x; idx < 64 * 128; idx += 256) {
        const int j = idx >> 7, n = idx & 127;
        float2 p;
        p.x = W[(2 * j + 0) * D + n];
        p.y = W[(2 * j + 1) * D + n];
        *(float2*)&wl[(j * BPITCH + n) * 2] = p;
    }
    __syncthreads();

    const int lane  = threadIdx.x & 31;
    const int wave  = threadIdx.x >> 5;
    const int strip = blockIdx.x * 8 + wave;
    const int row0  = strip * 16;
    if (row0 >= M) return;                 // wave-uniform: EXEC stays full for WMMA

    const int half = lane >> 4;            // 0: K pair {0,1}, 1: K pair {2,3}
    const int mr   = lane & 15;
    const int row  = row0 + mr;

    v8f zero = {};
    v8f acc[8];
#pragma unroll
    for (int t = 0; t < 8; ++t) acc[t] = zero;

    const float* bbase = &wl[(half * BPITCH + mr) * 2];

    if (row0 + 16 <= M) {
        // fast path: no per-lane guards in the hot loop
        const float* arow = X + (size_t)row * D + 2 * half;
        for (int k = 0; k < D; k += 4) {
            const float2 av = *(const float2*)(arow + k);
            v2f a; a.x = av.x; a.y = av.y;
            const float* bk = bbase + (k >> 1) * (BPITCH * 2);
            float2 bv[8];
#pragma unroll
            for (int t = 0; t < 8; ++t) bv[t] = *(const float2*)(bk + t * 32);
#pragma unroll
            for (int t = 0; t < 8; ++t) {
                v2f b; b.x = bv[t].x; b.y = bv[t].y;
                acc[t] = __builtin_amdgcn_wmma_f32_16x16x4_f32(
                    false, a, false, b, (short)0, acc[t], false, false);
            }
        }
    } else {
        // tail strip: branch-free per-lane masking (v_cndmask, EXEC untouched)
        const bool rowOK = row < M;
        const float* arow = X + (size_t)(rowOK ? row : row0) * D + 2 * half;
        for (int k = 0; k < D; k += 4) {
            const float2 av = *(const float2*)(arow + k);
            v2f a;
            a.x = rowOK ? av.x : 0.0f;
            a.y = rowOK ? av.y : 0.0f;
            const float* bk = bbase + (k >> 1) * (BPITCH * 2);
            float2 bv[8];
#pragma unroll
            for (int t = 0; t < 8; ++t) bv[t] = *(const float2*)(bk + t * 32);
#pragma unroll
            for (int t = 0; t < 8; ++t) {
                v2f b; b.x = bv[t].x; b.y = bv[t].y;
                acc[t] = __builtin_amdgcn_wmma_f32_16x16x4_f32(
                    false, a, false, b, (short)0, acc[t], false, false);
            }
        }
    }

    // C/D layout: VGPR r -> M = r + 8*half, N = mr
#pragma unroll
    for (int r = 0; r < 8; ++r) {
        const int orow = row0 + r + 8 * half;
        if (orow < M) {
#pragma unroll
            for (int t = 0; t < 8; ++t)
                OUT[(size_t)orow * D + t * 16 + mr] = acc[t][r];
        }
    }
}

// ---------------- aggregation ----------------
// G[n][:] = H[n][:] * dinv[n]^2   (self-loop term; also serves as zero-init)
__global__ void k_selfinit(const float* __restrict__ H, const float* __restrict__ dinv,
                           float* __restrict__ G, int N) {
    int i = blockIdx.x * blockDim.x + threadIdx.x;   // over N*32 float4 chunks
    int node = i >> 5, c = i & 31;
    if (node >= N) return;
    float w = dinv[node]; w *= w;
    float4 v = *(const float4*)(H + (size_t)node * D + c * 4);
    float4* g = (float4*)(G + (size_t)node * D + c * 4);
    *g = make_float4(v.x * w, v.y * w, v.z * w, v.w * w);
}

// Warp-per-edge scatter: lanes cover the 128-float row in float4 chunks.
// src/dst/norm uniform per wave; gather coalesced; scatter = L2 f32 atomics.
__global__ __launch_bounds__(256)
void k_scatter(const int* __restrict__ src, const int* __restrict__ dst,
               const float* __restrict__ dinv, const float* __restrict__ H,
               float* __restrict__ G, int E) {
    long long t = (long long)blockIdx.x * blockDim.x + threadIdx.x;
    int e = (int)(t >> 5);
    int c = (int)(t & 31);
    if (e >= E) return;
    const int s = src[e];
    const int d = dst[e];
    const float w = dinv[s] * dinv[d];
    float4 v = *(const float4*)(H + (size_t)s * D + c * 4);
    float* g = G + (size_t)d * D + c * 4;
    atomicAdd(g + 0, v.x * w);
    atomicAdd(g + 1, v.y * w);
    atomicAdd(g + 2, v.z * w);
    atomicAdd(g + 3, v.w * w);
}

// X = relu(G + b)
__global__ void k_bias_relu(const float* __restrict__ G, const float* __restrict__ b,
                            float* __restrict__ X, int total) {
    int i = blockIdx.x * blockDim.x + threadIdx.x;
    if (i >= total) return;
    float v = G[i] + b[i & (D - 1)];
    X[i] = v > 0.0f ? v : 0.0f;
}

// ---------------- head: sigmoid(X @ head_w + head_b) ----------------
__global__ void k_head(const float* __restrict__ X, const float* __restrict__ hw,
                       const float* __restrict__ hb, float* __restrict__ out, int N) {
    long long t = (long long)blockIdx.x * blockDim.x + threadIdx.x;
    int node = (int)(t >> 5);
    int lane = (int)(t & 31);
    if (node >= N) return;
    float4 x = *(const float4*)(X + (size_t)node * D + lane * 4);
    float4 w = *(const float4*)(hw + lane * 4);
    float s = x.x * w.x + x.y * w.y + x.z * w.z + x.w * w.w;
#pragma unroll
    for (int off = 16; off > 0; off >>= 1) s += __shfl_xor(s, off, 32);
    if (lane == 0) out[node] = 1.0f / (1.0f + __expf(-(s + hb[0])));
}

// ---------------- launch ----------------
extern "C" void kernel_launch(void* const* d_in, const int* in_sizes, int n_in,
                              void* d_out, int out_size, void* d_ws, size_t ws_size,
                              hipStream_t stream) {
    const int*   edge = (const int*)d_in[0];
    const float* emb  = (const float*)d_in[1];
    const float* W1   = (const float*)d_in[2];
    const float* b1   = (const float*)d_in[3];
    const float* W2   = (const float*)d_in[4];
    const float* b2   = (const float*)d_in[5];
    const float* hw   = (const float*)d_in[6];
    const float* hb   = (const float*)d_in[7];
    float* out = (float*)d_out;

    const int E = in_sizes[0] / 2;
    const int N = in_sizes[1] / D;
    const int* src = edge;
    const int* dst = edge + E;

    // workspace carve-up
    char* ws = (char*)d_ws;
    size_t off = 0;
    float* dinv = (float*)(ws + off); off += ((size_t)N * 4 + 255) & ~(size_t)255;
    float* H    = (float*)(ws + off); off += (size_t)N * D * 4;
    float* G    = (float*)(ws + off); off += (size_t)N * D * 4;
    float* X    = (float*)(ws + off); off += (size_t)N * D * 4;

    const int strips      = (N + 15) / 16;
    const int gemm_blocks = (strips + 7) / 8;
    const long long edgeThreads = (long long)E * 32;
    const int scat_blocks = (int)((edgeThreads + 255) / 256);
    const int nd          = N * D;

    // norm
    k_deg_init <<<(N + 255) / 256, 256, 0, stream>>>(dinv, N);
    k_deg_count<<<(E + 255) / 256, 256, 0, stream>>>(dst, dinv, E);
    k_rsqrt    <<<(N + 255) / 256, 256, 0, stream>>>(dinv, N);

    // layer 1
    k_gemm_wmma<<<gemm_blocks, 256, GEMM_LDS_BYTES, stream>>>(emb, W1, H, N);
    k_selfinit <<<(N * 32 + 255) / 256, 256, 0, stream>>>(H, dinv, G, N);
    k_scatter  <<<scat_blocks, 256, 0, stream>>>(src, dst, dinv, H, G, E);
    k_bias_relu<<<(nd + 255) / 256, 256, 0, stream>>>(G, b1, X, nd);

    // layer 2
    k_gemm_wmma<<<gemm_blocks, 256, GEMM_LDS_BYTES, stream>>>(X, W2, H, N);
    k_selfinit <<<(N * 32 + 255) / 256, 256, 0, stream>>>(H, dinv, G, N);
    k_scatter  <<<scat_blocks, 256, 0, stream>>>(src, dst, dinv, H, G, E);
    k_bias_relu<<<(nd + 255) / 256, 256, 0, stream>>>(G, b2, X, nd);

    // head
    k_head<<<(N * 32 + 255) / 256, 256, 0, stream>>>(X, hw, hb, out, N);
}